// MambaBlock_21620865368756
// MI455X (gfx1250) — compile-verified
//
#include <hip/hip_runtime.h>
#include <hip/hip_bf16.h>
#include <math.h>

// ---------------- problem constants (from reference) ----------------
#define BQ   32
#define SQ   500
#define INQ  16
#define DMQ  256
#define NLQ  2
#define DIQ  512            // d_inner
#define DSQ  16             // d_state
#define DCQ  4              // d_conv
#define DTRQ 16             // dt_rank
#define MROW (BQ * SQ)      // 16000 flattened rows (multiple of 64)

#define USE_ASYNC_LDS 1     // gfx1250 GLOBAL_LOAD_ASYNC_TO_LDS path

typedef float v2f __attribute__((ext_vector_type(2)));
typedef float v8f __attribute__((ext_vector_type(8)));

__device__ __forceinline__ float siluf(float v) {
    return v / (1.0f + __expf(-v));
}
__device__ __forceinline__ float softplusf(float v) {
    return v > 20.0f ? v : log1pf(__expf(v));
}

#if USE_ASYNC_LDS
// gfx1250 async global->LDS copy (ASYNCcnt-tracked), 16B per lane.
__device__ __forceinline__ void async_ld_b128(void* lds_ptr, const float* gptr) {
    // LDS offset is the low 32 bits of the generic address (flat aperture rule, ISA 10.2)
    unsigned lds_off = (unsigned)(unsigned long long)lds_ptr;
    asm volatile("global_load_async_to_lds_b128 %0, %1, off"
                 :: "v"(lds_off), "v"(gptr)
                 : "memory");
}
__device__ __forceinline__ void wait_async0() {
    asm volatile("s_wait_asynccnt 0x0" ::: "memory");
}
#endif

// ---------------------------------------------------------------
// fp32 WMMA GEMM:  C[M,N] = A[M,K] * W[N,K]^T  (+ epilogue)
// Block: 128 threads (4 waves) -> 64x64 output tile.
//   wave w owns rows [w*16, w*16+16), all 4 column tiles.
// K staged through LDS in chunks of 16, double buffered, async fills.
// LDS rows padded to 20 floats -> conflict-free lane-strided b64 reads.
// MODE: 0 none, 1 +bias[col], 2 +bias[col]+aux[(row%smod)*N+col],
//       3 softplus(v + bias[col])
// NGUARD: clamp W-row loads / guard stores when N % 64 != 0 (N=48 case)
// ---------------------------------------------------------------
#define KBLK 16
#define LDP  20   // padded LDS row stride (floats)

template <int MODE, bool NGUARD>
__global__ void gemm_wmma_f32(const float* __restrict__ A, int lda,
                              const float* __restrict__ W,
                              const float* __restrict__ bias,
                              const float* __restrict__ aux,
                              float* __restrict__ C,
                              int M, int N, int K, int smod) {
    __shared__ float At[2][64 * LDP];
    __shared__ float Wt[2][64 * LDP];

    const int lane = threadIdx.x & 31;
    const int wave = threadIdx.x >> 5;
    const int mb = blockIdx.x * 64;            // block row base
    const int n0 = blockIdx.y * 64;            // block col base
    const int l  = lane & 15;
    const int hi = lane >> 4;

    v8f acc[4];
#pragma unroll
    for (int t = 0; t < 4; ++t) acc[t] = (v8f)(0.0f);

    const int nc = K / KBLK;

    // ---- tile fill: 64 rows x 16 floats = 256 float4, 128 threads ----
    auto fill = [&](int p, int kk) {
#pragma unroll
        for (int i = threadIdx.x; i < 256; i += 128) {
            const int row = i >> 2;
            const int kq  = (i & 3) << 2;
            const float* ga = A + (size_t)(mb + row) * lda + kk + kq;
            int wr = n0 + row;
            if (NGUARD) wr = wr < N ? wr : N - 1;
            const float* gw = W + (size_t)wr * K + kk + kq;
#if USE_ASYNC_LDS
            async_ld_b128(&At[p][row * LDP + kq], ga);
            async_ld_b128(&Wt[p][row * LDP + kq], gw);
#else
            *(float4*)&At[p][row * LDP + kq] = *(const float4*)ga;
            *(float4*)&Wt[p][row * LDP + kq] = *(const float4*)gw;
#endif
        }
    };

    fill(0, 0);
    int p = 0;
    for (int c = 0; c < nc; ++c) {
#if USE_ASYNC_LDS
        wait_async0();
#endif
        __syncthreads();                       // tile p complete for all waves
        if (c + 1 < nc) fill(p ^ 1, (c + 1) * KBLK);

        const float* at = &At[p][0];
        const float* wt = &Wt[p][0];
        const int ar = (wave * 16 + l) * LDP + 2 * hi;
#pragma unroll
        for (int s = 0; s < 4; ++s) {
            v2f av;
            av.x = at[ar + 4 * s + 0];
            av.y = at[ar + 4 * s + 1];
#pragma unroll
            for (int t = 0; t < 4; ++t) {
                const int wr = (t * 16 + l) * LDP + 4 * s + 2 * hi;
                v2f bv;
                bv.x = wt[wr + 0];
                bv.y = wt[wr + 1];
                acc[t] = __builtin_amdgcn_wmma_f32_16x16x4_f32(
                    false, av, false, bv, (short)0, acc[t], false, false);
            }
        }
        __syncthreads();                       // all reads of tile p done
        p ^= 1;
    }

#pragma unroll
    for (int t = 0; t < 4; ++t) {
        const int col = n0 + t * 16 + l;
        if (NGUARD && col >= N) continue;      // after all WMMAs: safe
#pragma unroll
        for (int r = 0; r < 8; ++r) {
            const int row = mb + wave * 16 + r + 8 * hi;
            float v = acc[t][r];
            if (MODE == 1) {
                v += bias[col];
            } else if (MODE == 2) {
                v += bias[col] + aux[(size_t)(row % smod) * N + col];
            } else if (MODE == 3) {
                v = softplusf(v + bias[col]);
            }
            C[(size_t)row * N + col] = v;
        }
    }
}

// ---------------------------------------------------------------
// Depthwise causal conv1d (kernel 4) + bias + SiLU.
// xz is [B*S, 2*DI]; xi is its first DI columns. Output xc [B*S, DI].
// ---------------------------------------------------------------
__global__ void conv_silu_kernel(const float* __restrict__ xz,
                                 const float* __restrict__ cw,
                                 const float* __restrict__ cb,
                                 float* __restrict__ xc) {
    const int idx = blockIdx.x * blockDim.x + threadIdx.x;  // over B*S*DI
    if (idx >= MROW * DIQ) return;
    const int d   = idx & (DIQ - 1);
    const int row = idx >> 9;            // /DIQ
    const int l   = row % SQ;
    const int b   = row / SQ;
    float acc = cb[d];
#pragma unroll
    for (int k = 0; k < DCQ; ++k) {
        const int ls = l + k - (DCQ - 1);
        if (ls >= 0) {
            acc += xz[(size_t)(b * SQ + ls) * (2 * DIQ) + d] * cw[d * DCQ + k];
        }
    }
    xc[idx] = siluf(acc);
}

// ---------------------------------------------------------------
// Selective scan. One block = (one batch, 256 channels); h[16] in regs.
// B_t / C_t (16 floats each) staged through LDS per timestep.
// ---------------------------------------------------------------
__global__ void scan_kernel(const float* __restrict__ xdbl,  // [B*S,48]
                            const float* __restrict__ dt,    // [B*S,DI]
                            const float* __restrict__ xc,    // [B*S,DI]
                            const float* __restrict__ A_log, // [DI,DS]
                            float* __restrict__ y) {         // [B*S,DI]
    const int d = blockIdx.x * blockDim.x + threadIdx.x;     // channel
    const int b = blockIdx.y;

    float Arow[DSQ];
#pragma unroll
    for (int s = 0; s < DSQ; ++s) Arow[s] = -__expf(A_log[d * DSQ + s]);

    float h[DSQ];
#pragma unroll
    for (int s = 0; s < DSQ; ++s) h[s] = 0.0f;

    __shared__ float sB[DSQ];
    __shared__ float sC[DSQ];

    for (int l = 0; l < SQ; ++l) {
        const int row = b * SQ + l;
        if (threadIdx.x < DSQ) {
            sB[threadIdx.x] = xdbl[(size_t)row * 48 + DTRQ + threadIdx.x];
        } else if (threadIdx.x < 2 * DSQ) {
            sC[threadIdx.x - DSQ] =
                xdbl[(size_t)row * 48 + DTRQ + DSQ + (threadIdx.x - DSQ)];
        }
        __syncthreads();

        const float dtv = dt[(size_t)row * DIQ + d];
        const float xv  = xc[(size_t)row * DIQ + d];
        const float dx  = dtv * xv;
        float accy = 0.0f;
#pragma unroll
        for (int s = 0; s < DSQ; ++s) {
            const float dA = __expf(dtv * Arow[s]);
            h[s] = dA * h[s] + dx * sB[s];
            accy += h[s] * sC[s];
        }
        y[(size_t)row * DIQ + d] = accy;
        __syncthreads();
    }
}

// ---------------------------------------------------------------
// y2 = (y + Dp[d]*xc) * silu(z),  z = xz[:, DI + d]
// ---------------------------------------------------------------
__global__ void gate_kernel(const float* __restrict__ y,
                            const float* __restrict__ xc,
                            const float* __restrict__ xz,
                            const float* __restrict__ Dp,
                            float* __restrict__ y2) {
    const int idx = blockIdx.x * blockDim.x + threadIdx.x;  // over B*S*DI
    if (idx >= MROW * DIQ) return;
    const int d   = idx & (DIQ - 1);
    const int row = idx >> 9;
    const float z = xz[(size_t)row * (2 * DIQ) + DIQ + d];
    const float v = y[idx] + Dp[d] * xc[idx];
    y2[idx] = v * siluf(z);
}

// ---------------------------------------------------------------
extern "C" void kernel_launch(void* const* d_in, const int* in_sizes, int n_in,
                              void* d_out, int out_size, void* d_ws, size_t ws_size,
                              hipStream_t stream) {
    (void)in_sizes; (void)n_in; (void)out_size; (void)ws_size;

    const float* x        = (const float*)d_in[0];   // [B,S,16]
    const float* embed_w  = (const float*)d_in[1];   // [256,16]
    const float* embed_b  = (const float*)d_in[2];   // [256]
    const float* pos_emb  = (const float*)d_in[3];   // [S,256]
    const float* in_w     = (const float*)d_in[4];   // [NL,1024,256]
    const float* conv_w   = (const float*)d_in[5];   // [NL,512,4]
    const float* conv_b   = (const float*)d_in[6];   // [NL,512]
    const float* xp_w     = (const float*)d_in[7];   // [NL,48,512]
    const float* dt_w     = (const float*)d_in[8];   // [NL,512,16]
    const float* dt_b     = (const float*)d_in[9];   // [NL,512]
    const float* A_log    = (const float*)d_in[10];  // [NL,512,16]
    const float* Dp       = (const float*)d_in[11];  // [NL,512]
    const float* out_w    = (const float*)d_in[12];  // [NL,256,512]
    const float* proj_w   = (const float*)d_in[13];  // [256,256]
    const float* proj_b   = (const float*)d_in[14];  // [256]
    float* out = (float*)d_out;                      // [B,S,256] fp32

    // workspace layout (floats)
    float* ws = (float*)d_ws;
    size_t o = 0;
    float* hbuf  = ws + o; o += (size_t)MROW * DMQ;        // [16000,256]
    float* xzbuf = ws + o; o += (size_t)MROW * 2 * DIQ;    // [16000,1024]
    float* xcbuf = ws + o; o += (size_t)MROW * DIQ;        // [16000,512]
    float* xdbl  = ws + o; o += (size_t)MROW * 48;         // [16000,48]
    float* dtbuf = ws + o; o += (size_t)MROW * DIQ;        // [16000,512] (reused as y2)
    float* ybuf  = ws + o; o += (size_t)MROW * DIQ;        // [16000,512]

    const dim3 blk(128);                       // 4 waves of 32
    const int  mgrid = MROW / 64;              // 250 M-blocks
    const int  ew = (MROW * DIQ + 255) / 256;  // elementwise grid

    // 1) embed: h = x @ embed_w^T + embed_b + pos_emb[l]
    gemm_wmma_f32<2, false><<<dim3(mgrid, DMQ / 64), blk, 0, stream>>>(
        x, INQ, embed_w, embed_b, pos_emb, hbuf, MROW, DMQ, INQ, SQ);

    for (int layer = 0; layer < NLQ; ++layer) {
        const float* iw  = in_w   + (size_t)layer * 2 * DIQ * DMQ;
        const float* cwl = conv_w + (size_t)layer * DIQ * DCQ;
        const float* cbl = conv_b + (size_t)layer * DIQ;
        const float* xpl = xp_w   + (size_t)layer * 48 * DIQ;
        const float* dwl = dt_w   + (size_t)layer * DIQ * DTRQ;
        const float* dbl_ = dt_b  + (size_t)layer * DIQ;
        const float* Al  = A_log  + (size_t)layer * DIQ * DSQ;
        const float* Dl  = Dp     + (size_t)layer * DIQ;
        const float* owl = out_w  + (size_t)layer * DMQ * DIQ;

        // 2a) in_proj: xz = h @ in_w^T   [16000,1024]
        gemm_wmma_f32<0, false><<<dim3(mgrid, (2 * DIQ) / 64), blk, 0, stream>>>(
            hbuf, DMQ, iw, nullptr, nullptr, xzbuf, MROW, 2 * DIQ, DMQ, 0);

        // 2b) causal depthwise conv + SiLU -> xc
        conv_silu_kernel<<<ew, 256, 0, stream>>>(xzbuf, cwl, cbl, xcbuf);

        // 2c) x_proj: xdbl = xc @ xp_w^T  [16000,48]  (N=48 -> guarded)
        gemm_wmma_f32<0, true><<<dim3(mgrid, 1), blk, 0, stream>>>(
            xcbuf, DIQ, xpl, nullptr, nullptr, xdbl, MROW, 48, DIQ, 0);

        // 2d) dt = softplus(dt_cols(xdbl) @ dt_w^T + dt_b)  [16000,512]
        gemm_wmma_f32<3, false><<<dim3(mgrid, DIQ / 64), blk, 0, stream>>>(
            xdbl, 48, dwl, dbl_, nullptr, dtbuf, MROW, DIQ, DTRQ, 0);

        // 2e) selective scan -> y
        scan_kernel<<<dim3(DIQ / 256, BQ), 256, 0, stream>>>(
            xdbl, dtbuf, xcbuf, Al, ybuf);

        // 2f) gating: y2 = (y + Dp*xc) * silu(z)   (overwrites dtbuf)
        gate_kernel<<<ew, 256, 0, stream>>>(ybuf, xcbuf, xzbuf, Dl, dtbuf);

        // 2g) out_proj: h = y2 @ out_w^T  [16000,256]
        gemm_wmma_f32<0, false><<<dim3(mgrid, DMQ / 64), blk, 0, stream>>>(
            dtbuf, DIQ, owl, nullptr, nullptr, hbuf, MROW, DMQ, DIQ, 0);
    }

    // 3) final projection: out = h @ proj_w^T + proj_b
    gemm_wmma_f32<1, false><<<dim3(mgrid, DMQ / 64), blk, 0, stream>>>(
        hbuf, DMQ, proj_w, proj_b, nullptr, out, MROW, DMQ, DMQ, 0);
}